// Density_prop_CNN_31997506355862
// MI455X (gfx1250) — compile-verified
//
#include <hip/hip_runtime.h>
#include <math.h>

typedef __attribute__((ext_vector_type(2))) float v2f;
typedef __attribute__((ext_vector_type(8))) float v8f;

#define Bn 16
#define Kc 16
#define P24 24
#define P 576
#define Po 144
#define U 10

__device__ __forceinline__ float softplusf(float x) { return log1pf(expf(x)); }
__device__ __forceinline__ float cleanf(float v) { return (isnan(v) || isinf(v)) ? 0.f : v; }

// ---------------------------------------------------------------------------
// Kernel A: conv5x5 -> relu/maxpool argmax -> gather gated patches Y, mu3,
// idx, g, and per-(b,k) trace partial  sp_c[k] * sum_p ||Y_p||^2.
// grid = 256 (b*16+k), block = 128
// ---------------------------------------------------------------------------
__global__ void convpool_kernel(const float* __restrict__ in,
                                const float* __restrict__ wconv,
                                const float* __restrict__ wsc,
                                float* __restrict__ Y,      // [256][144][32]
                                float* __restrict__ mu3,    // [B][144][16] (h,w,c order)
                                int*   __restrict__ idx,    // [256][144]
                                float* __restrict__ gbuf,   // [256][144]
                                float* __restrict__ trp)    // [256]
{
    const int bk = blockIdx.x, b = bk >> 4, k = bk & 15;
    __shared__ float img[784];
    __shared__ float wk[25];
    __shared__ float mu1[P];
    __shared__ float red[128];
    const int t = threadIdx.x;

    for (int i = t; i < 784; i += 128) img[i] = in[b * 784 + i];
    if (t < 25) wk[t] = wconv[t * Kc + k];   // [5,5,1,16] -> (di*5+dj)*16+k
    __syncthreads();

    for (int p = t; p < P; p += 128) {
        const int y = p / P24, x = p % P24;
        float s = 0.f;
#pragma unroll
        for (int di = 0; di < 5; ++di)
#pragma unroll
            for (int dj = 0; dj < 5; ++dj)
                s += wk[di * 5 + dj] * img[(y + di) * 28 + (x + dj)];
        mu1[p] = s;
    }
    __syncthreads();

    const float sp = softplusf(wsc[k]);
    float trpart = 0.f;
    for (int po = t; po < Po; po += 128) {
        const int oy = po / 12, ox = po % 12;
        float best = 0.f; int bj = 0;
#pragma unroll
        for (int j = 0; j < 4; ++j) {     // jnp.argmax: first occurrence wins
            const int yy = 2 * oy + (j >> 1), xx = 2 * ox + (j & 1);
            float v = mu1[yy * P24 + xx]; v = v > 0.f ? v : 0.f;
            if (j == 0) { best = v; bj = 0; }
            else if (v > best) { best = v; bj = j; }
        }
        const int yy = 2 * oy + (bj >> 1), xx = 2 * ox + (bj & 1);
        const int pi = yy * P24 + xx;
        const float g = (mu1[pi] > 0.f) ? 1.f : 0.f;
        mu3[b * (Po * Kc) + po * Kc + k] = best;
        idx[bk * Po + po] = pi;
        gbuf[bk * Po + po] = g;
        float* yr = Y + ((size_t)bk * Po + po) * 32;
        float nrm = 0.f;
#pragma unroll
        for (int di = 0; di < 5; ++di)
#pragma unroll
            for (int dj = 0; dj < 5; ++dj) {
                const float v = g * img[(yy + di) * 28 + (xx + dj)];
                yr[di * 5 + dj] = v; nrm += v * v;
            }
#pragma unroll
        for (int c = 25; c < 32; ++c) yr[c] = 0.f;
        trpart += nrm;
    }
    red[t] = trpart; __syncthreads();
    for (int s2 = 64; s2 > 0; s2 >>= 1) {
        if (t < s2) red[t] += red[t + s2];
        __syncthreads();
    }
    if (t == 0) trp[bk] = sp * red[0];
}

// ---------------------------------------------------------------------------
// Kernel B: per (b,k) GEMM  Z[16,32] = Wc_k^T[16x144] * (sqrt(sp_c)*Y)[144x32]
// via V_WMMA_F32_16X16X4_F32.  grid = 256, block = 32 (one wave).
// ---------------------------------------------------------------------------
__global__ void gemm1_kernel(const float* __restrict__ wfc,
                             const float* __restrict__ wsc,
                             const float* __restrict__ Y,
                             float* __restrict__ Z)       // [256][16][32]
{
    const int bk = blockIdx.x, k = bk & 15;
    __shared__ float W[Po * 16];                   // [row][u], u>=10 zero-pad
    const int t = threadIdx.x;
    for (int s = t; s < Po * 16; s += 32) {
        const int row = s >> 4, u = s & 15;
        W[s] = (u < U) ? wfc[(k * Po + row) * U + u] : 0.f;
    }
    __syncthreads();
    const float ssc = sqrtf(softplusf(wsc[k]));
    const int half = t >> 4, lo = t & 15;
    v8f acc0 = {}; v8f acc1 = {};
    const float* Yb = Y + (size_t)bk * Po * 32;
    for (int step = 0; step < 36; ++step) {
        const int r0 = 4 * step + 2 * half, r1 = r0 + 1;
        v2f av; av.x = W[r0 * 16 + lo];           av.y = W[r1 * 16 + lo];
        v2f b0; b0.x = ssc * Yb[r0 * 32 + lo];    b0.y = ssc * Yb[r1 * 32 + lo];
        v2f b1; b1.x = ssc * Yb[r0 * 32 + 16 + lo]; b1.y = ssc * Yb[r1 * 32 + 16 + lo];
        acc0 = __builtin_amdgcn_wmma_f32_16x16x4_f32(false, av, false, b0, (short)0, acc0, false, false);
        acc1 = __builtin_amdgcn_wmma_f32_16x16x4_f32(false, av, false, b1, (short)0, acc1, false, false);
    }
    float* Zb = Z + (size_t)bk * 16 * 32;
#pragma unroll
    for (int j = 0; j < 8; ++j) {
        const int u = j + 8 * half;
        Zb[u * 32 + lo]      = acc0[j];
        Zb[u * 32 + 16 + lo] = acc1[j];
    }
}

// ---------------------------------------------------------------------------
// Kernel C: per b — symmetric WMMA GEMM Sigma_a = sum_k Z_k Z_k^T, set_diag
// quirk correction (rank-1), diag_extra, softmax, grad sandwich, outputs.
// grid = 16, block = 32 (one wave).
// ---------------------------------------------------------------------------
__global__ void finalize_kernel(const float* __restrict__ in,
                                const float* __restrict__ wfc,
                                const float* __restrict__ wsc,
                                const float* __restrict__ wsf,
                                const float* __restrict__ Y,
                                const float* __restrict__ Z,
                                const float* __restrict__ mu3,
                                const int*   __restrict__ idx,
                                const float* __restrict__ gbuf,
                                const float* __restrict__ trp,
                                float* __restrict__ out)
{
    const int b = blockIdx.x, t = threadIdx.x;
    const int half = t >> 4, lo = t & 15;
    __shared__ float Sig[256];
    __shared__ float xmk[32];
    __shared__ float dsb[Po];
    __shared__ float mu4s[16];
    __shared__ float ps[16];
    __shared__ float redm[32];

    // Sigma_a = sum_k Zs_k Zs_k^T : A and B per-lane data are identical.
    v8f acc = {};
    for (int k = 0; k < Kc; ++k) {
        const float* Zb = Z + ((size_t)(b * Kc + k) * 16 + lo) * 32;
#pragma unroll
        for (int step = 0; step < 7; ++step) {
            const int c0 = 4 * step + 2 * half;
            v2f z; z.x = Zb[c0]; z.y = Zb[c0 + 1];
            acc = __builtin_amdgcn_wmma_f32_16x16x4_f32(false, z, false, z, (short)0, acc, false, false);
        }
    }
#pragma unroll
    for (int j = 0; j < 8; ++j) Sig[(j + 8 * half) * 16 + lo] = acc[j];
    __syncthreads();

    // set_diag quirk correction: only column qo=k/2 when argmax landed on
    // spatial index k.  Delta = sp*2*max(0,-Y_p.xm_k) (times g_k).
    for (int k = 0; k < Kc; ++k) {
        const int qo = k >> 1;
        const int pisel = idx[(b * Kc + k) * Po + qo];
        const float gk = gbuf[(b * Kc + k) * Po + qo];
        if (pisel == k && gk != 0.f) {            // uniform branch
            if (t < 25) { const int di = t / 5, dj = t % 5; xmk[t] = in[b * 784 + di * 28 + (k + dj)]; }
            __syncthreads();
            const float sp = softplusf(wsc[k]);
            const float* Yb = Y + (size_t)(b * Kc + k) * Po * 32;
            for (int p = t; p < Po; p += 32) {
                float dot = 0.f;
#pragma unroll
                for (int c = 0; c < 25; ++c) dot += Yb[p * 32 + c] * xmk[c];
                float d = -dot; d = d > 0.f ? d : 0.f;
                dsb[p] = sp * 2.f * d;
            }
            __syncthreads();
            if (t < U) {
                float dv = 0.f;
                for (int p = 0; p < Po; ++p) dv += dsb[p] * wfc[(k * Po + p) * U + t];
                for (int v = 0; v < U; ++v) Sig[t * 16 + v] += dv * wfc[(k * Po + qo) * U + v];
            }
            __syncthreads();
        }
    }

    // ||mu_flat||^2 and mu4
    float part = 0.f;
    for (int j = t; j < Po * Kc; j += 32) { const float f = mu3[b * (Po * Kc) + j]; part += f * f; }
    redm[t] = part; __syncthreads();
    if (t < 16) redm[t] += redm[t + 16]; __syncthreads();
    if (t < 8)  redm[t] += redm[t + 8];  __syncthreads();
    if (t < 4)  redm[t] += redm[t + 4];  __syncthreads();
    if (t < 2)  redm[t] += redm[t + 2];  __syncthreads();
    if (t == 0) redm[0] += redm[1];      __syncthreads();
    const float msq = redm[0];
    if (t < U) {
        float m4 = 0.f;
        for (int j = 0; j < Po * Kc; ++j) m4 += mu3[b * (Po * Kc) + j] * wfc[j * U + t];
        mu4s[t] = m4;
    }
    __syncthreads();

    float tr = 0.f;
    for (int k = 0; k < Kc; ++k) tr += trp[b * Kc + k];   // fixed order: deterministic

    if (t < U) Sig[t * 16 + t] += softplusf(wsf[t]) * (tr + msq);
    __syncthreads();
    for (int e = t; e < 100; e += 32) {                   // clean + abs diag
        const int i = e / 10, j2 = e % 10;
        float v = cleanf(Sig[i * 16 + j2]);
        if (i == j2) v = fabsf(v);
        Sig[i * 16 + j2] = v;
    }
    __syncthreads();

    if (t < U) {                                          // softmax
        float mx = mu4s[0];
        for (int u = 1; u < U; ++u) mx = fmaxf(mx, mu4s[u]);
        float den = 0.f;
        for (int u = 0; u < U; ++u) den += expf(mu4s[u] - mx);
        ps[t] = expf(mu4s[t] - mx) / den;
    }
    __syncthreads();
    if (t < U) out[b * U + t] = ps[t];

    for (int e = t; e < 100; e += 32) {                   // Sout = G S4 G^T
        const int i = e / 10, j2 = e % 10;
        float s = 0.f;
        for (int m = 0; m < U; ++m) {
            const float Gim = (i == m ? ps[i] : 0.f) - ps[i] * ps[m];
            float tmj = 0.f;
            for (int l = 0; l < U; ++l) {
                const float Gjl = (j2 == l ? ps[j2] : 0.f) - ps[j2] * ps[l];
                tmj += Sig[m * 16 + l] * Gjl;
            }
            s += Gim * tmj;
        }
        s = cleanf(s);
        if (i == j2) s = fabsf(s);
        s = cleanf(s);
        out[Bn * U + b * 100 + i * 10 + j2] = s;
    }
}

extern "C" void kernel_launch(void* const* d_in, const int* in_sizes, int n_in,
                              void* d_out, int out_size, void* d_ws, size_t ws_size,
                              hipStream_t stream) {
    (void)in_sizes; (void)n_in; (void)out_size; (void)ws_size;
    const float* d_inputs = (const float*)d_in[0];
    const float* d_wconv  = (const float*)d_in[1];
    const float* d_wsc    = (const float*)d_in[2];
    const float* d_wfc    = (const float*)d_in[3];
    const float* d_wsf    = (const float*)d_in[4];
    float* out = (float*)d_out;

    float* ws  = (float*)d_ws;
    float* Y   = ws;                 // 256*144*32      = 1,179,648 f
    float* Z   = Y   + 1179648;      // 256*16*32       =   131,072 f
    float* m3  = Z   + 131072;       // 16*144*16       =    36,864 f
    float* gb  = m3  + 36864;        // 256*144         =    36,864 f
    float* trp = gb  + 36864;        // 256             f
    int*   idx = (int*)(trp + 256);  // 256*144         i

    convpool_kernel<<<256, 128, 0, stream>>>(d_inputs, d_wconv, d_wsc, Y, m3, idx, gb, trp);
    gemm1_kernel  <<<256,  32, 0, stream>>>(d_wfc, d_wsc, Y, Z);
    finalize_kernel<<<16,  32, 0, stream>>>(d_inputs, d_wfc, d_wsc, d_wsf, Y, Z, m3, idx, gb, trp, out);
}